// LinearAttention_1666447311143
// MI455X (gfx1250) — compile-verified
//
#include <hip/hip_runtime.h>

typedef __attribute__((ext_vector_type(2))) float v2f;
typedef __attribute__((ext_vector_type(8))) float v8f;

#define HEADS    4
#define DIM_HEAD 32
#define CIN      256          // input channels
#define HID      128          // HEADS*DIM_HEAD
#define O3       384          // 3*HID
#define NPIX     4096         // 64*64
#define NB       16           // batch

// ---------------------------------------------------------------------------
// Kernel 1: qkv[b] (384 x 4096) = w_qkv (384 x 256) @ x[b] (256 x 4096)
// One wave computes a 16(M) x 64(N) tile: A (16x4) reused across 4 B tiles.
// f32 WMMA operand layout (ISA 7.12.2):
//   A: lane<16 -> M=lane, K={kk,kk+1};  lane>=16 -> M=lane-16, K={kk+2,kk+3}
//   B: lane<16 -> N=lane, K={kk,kk+1};  lane>=16 -> N=lane-16, K={kk+2,kk+3}
//   C/D: VGPR v -> M = v + (lane<16 ? 0 : 8), N = lane&15
// ---------------------------------------------------------------------------
__global__ void __launch_bounds__(256) qkv_gemm(const float* __restrict__ x,
                                                const float* __restrict__ w_qkv,
                                                float* __restrict__ qkv) {
    const int wave = (int)((blockIdx.x * blockDim.x + threadIdx.x) >> 5);
    const int lane = (int)(threadIdx.x & 31);
    const int ln   = lane & 15;
    const int hi   = lane >> 4;          // 0 or 1
    const int koff = hi * 2;

    const int b  = wave / (24 * 64);     // 24 M-tiles, 64 N-groups of 64
    const int r  = wave % (24 * 64);
    const int mt = r >> 6;
    const int ng = r & 63;
    const int m0 = mt * 16;
    const int n0 = ng * 64;

    const float* __restrict__ wrow = w_qkv + (size_t)(m0 + ln) * CIN;
    const float* __restrict__ xb   = x + (size_t)b * CIN * NPIX;

    v8f acc0 = {}, acc1 = {}, acc2 = {}, acc3 = {};
    for (int kk = 0; kk < CIN; kk += 4) {
        v2f A = *(const v2f*)(wrow + kk + koff);                 // 8B aligned
        const float* __restrict__ xk = xb + (size_t)(kk + koff) * NPIX + n0 + ln;
        v2f B0 = { xk[0],  xk[0 + NPIX]  };
        v2f B1 = { xk[16], xk[16 + NPIX] };
        v2f B2 = { xk[32], xk[32 + NPIX] };
        v2f B3 = { xk[48], xk[48 + NPIX] };
        acc0 = __builtin_amdgcn_wmma_f32_16x16x4_f32(false, A, false, B0, (short)0, acc0, false, false);
        acc1 = __builtin_amdgcn_wmma_f32_16x16x4_f32(false, A, false, B1, (short)0, acc1, false, false);
        acc2 = __builtin_amdgcn_wmma_f32_16x16x4_f32(false, A, false, B2, (short)0, acc2, false, false);
        acc3 = __builtin_amdgcn_wmma_f32_16x16x4_f32(false, A, false, B3, (short)0, acc3, false, false);
    }

    float* __restrict__ qb = qkv + ((size_t)b * O3 + m0) * NPIX;
    const int rh = hi * 8;
#pragma unroll
    for (int v = 0; v < 8; ++v) {
        float* dst = qb + (size_t)(v + rh) * NPIX + n0 + ln;
        dst[0]  = acc0[v];
        dst[16] = acc1[v];
        dst[32] = acc2[v];
        dst[48] = acc3[v];
    }
}

// ---------------------------------------------------------------------------
// Kernel 2: per-row softmax statistics for the K part (rows 128..255 of qkv).
// 2048 rows (b*128 + h*32 + d), one 256-thread block per row.
// ---------------------------------------------------------------------------
__global__ void __launch_bounds__(256) softmax_stats(const float* __restrict__ qkv,
                                                     float* __restrict__ rmax,
                                                     float* __restrict__ rsum) {
    const int row = (int)blockIdx.x;            // 0..2047
    const int b   = row >> 7;
    const int loc = row & 127;
    const float* __restrict__ kr = qkv + ((size_t)b * O3 + HID + loc) * NPIX;

    __shared__ float red[256];
    float m = -3.402823466e38f;
    for (int i = (int)threadIdx.x; i < NPIX; i += 256) m = fmaxf(m, kr[i]);
    red[threadIdx.x] = m;
    __syncthreads();
    for (int s = 128; s > 0; s >>= 1) {
        if ((int)threadIdx.x < s) red[threadIdx.x] = fmaxf(red[threadIdx.x], red[threadIdx.x + s]);
        __syncthreads();
    }
    m = red[0];
    __syncthreads();

    float sum = 0.0f;
    for (int i = (int)threadIdx.x; i < NPIX; i += 256) sum += __expf(kr[i] - m);
    red[threadIdx.x] = sum;
    __syncthreads();
    for (int s = 128; s > 0; s >>= 1) {
        if ((int)threadIdx.x < s) red[threadIdx.x] += red[threadIdx.x + s];
        __syncthreads();
    }
    if (threadIdx.x == 0) { rmax[row] = m; rsum[row] = red[0]; }
}

// ---------------------------------------------------------------------------
// Kernel 3: ctx[b,h](32x32) = softmax(K)(32x4096) @ V(32x4096)^T
// One wave per (b,h,dtile): 16(d) x 32(e) output, K-loop over n.
// A = exp(k - rowmax) computed on the fly; normalize by 1/rowsum at store.
// ---------------------------------------------------------------------------
__global__ void __launch_bounds__(32) context_wmma(const float* __restrict__ qkv,
                                                   const float* __restrict__ rmax,
                                                   const float* __restrict__ rsum,
                                                   float* __restrict__ ctx) {
    const int wave = (int)blockIdx.x;           // 0..127 = b*8 + h*2 + dt
    const int b  = wave >> 3;
    const int h  = (wave >> 1) & 3;
    const int dt = wave & 1;
    const int lane = (int)(threadIdx.x & 31);
    const int ln = lane & 15;
    const int hi = lane >> 4;
    const int koff = hi * 2;
    const int d0 = dt * 16;

    const float* __restrict__ kd = qkv + ((size_t)b * O3 + HID + h * 32 + d0 + ln) * NPIX;
    const float* __restrict__ vb = qkv + ((size_t)b * O3 + 2 * HID + h * 32) * NPIX;
    const float md = rmax[b * 128 + h * 32 + d0 + ln];

    v8f acc0 = {}, acc1 = {};
    for (int kk = 0; kk < NPIX; kk += 4) {
        v2f kv = *(const v2f*)(kd + kk + koff);
        v2f A  = { __expf(kv.x - md), __expf(kv.y - md) };
        v2f B0 = *(const v2f*)(vb + (size_t)ln * NPIX + kk + koff);         // e = ln
        v2f B1 = *(const v2f*)(vb + (size_t)(16 + ln) * NPIX + kk + koff);  // e = 16+ln
        acc0 = __builtin_amdgcn_wmma_f32_16x16x4_f32(false, A, false, B0, (short)0, acc0, false, false);
        acc1 = __builtin_amdgcn_wmma_f32_16x16x4_f32(false, A, false, B1, (short)0, acc1, false, false);
    }

    float* __restrict__ cb = ctx + (size_t)(b * 4 + h) * 32 * 32;
#pragma unroll
    for (int v = 0; v < 8; ++v) {
        const int d = d0 + v + hi * 8;
        const float inv = 1.0f / rsum[b * 128 + h * 32 + d];
        cb[(size_t)d * 32 + ln]      = acc0[v] * inv;
        cb[(size_t)d * 32 + 16 + ln] = acc1[v] * inv;
    }
}

// ---------------------------------------------------------------------------
// Kernel 4: W_eff[b][o][h*32+d] = sum_e w_out[o][h*32+e] * ctx[b,h,d,e]
// Tiny (16.7 MFLOP) — plain VALU.
// ---------------------------------------------------------------------------
__global__ void __launch_bounds__(256) build_weff(const float* __restrict__ w_out,
                                                  const float* __restrict__ ctx,
                                                  float* __restrict__ weff) {
    const int idx = (int)(blockIdx.x * 256 + threadIdx.x);   // < 16*256*128
    const int hd = idx & 127;
    const int o  = (idx >> 7) & 255;
    const int b  = idx >> 15;
    const int h  = hd >> 5;
    const int d  = hd & 31;
    const float* __restrict__ wr = w_out + (size_t)o * HID + h * 32;
    const float* __restrict__ cr = ctx + ((size_t)(b * 4 + h) * 32 + d) * 32;
    float s = 0.0f;
#pragma unroll
    for (int e = 0; e < 32; ++e) s += wr[e] * cr[e];
    weff[idx] = s;
}

// ---------------------------------------------------------------------------
// Kernel 5: y[b] (256 x 4096) = W_eff[b] (256 x 128) @ q[b] (128 x 4096) + b_out
// Same WMMA tiling as kernel 1 (16x64 tile per wave), bias fused at store.
// ---------------------------------------------------------------------------
__global__ void __launch_bounds__(256) out_gemm(const float* __restrict__ qkv,
                                                const float* __restrict__ weff,
                                                const float* __restrict__ b_out,
                                                float* __restrict__ y) {
    const int wave = (int)((blockIdx.x * blockDim.x + threadIdx.x) >> 5);
    const int lane = (int)(threadIdx.x & 31);
    const int ln   = lane & 15;
    const int hi   = lane >> 4;
    const int koff = hi * 2;

    const int b  = wave >> 10;                 // 16 M-tiles * 64 N-groups = 1024
    const int r  = wave & 1023;
    const int mt = r >> 6;
    const int ng = r & 63;
    const int m0 = mt * 16;
    const int n0 = ng * 64;

    const float* __restrict__ wrow = weff + ((size_t)b * 256 + m0 + ln) * HID;
    const float* __restrict__ qb   = qkv + (size_t)b * O3 * NPIX;   // q = rows 0..127

    v8f acc0 = {}, acc1 = {}, acc2 = {}, acc3 = {};
    for (int kk = 0; kk < HID; kk += 4) {
        v2f A = *(const v2f*)(wrow + kk + koff);
        const float* __restrict__ qk = qb + (size_t)(kk + koff) * NPIX + n0 + ln;
        v2f B0 = { qk[0],  qk[0 + NPIX]  };
        v2f B1 = { qk[16], qk[16 + NPIX] };
        v2f B2 = { qk[32], qk[32 + NPIX] };
        v2f B3 = { qk[48], qk[48 + NPIX] };
        acc0 = __builtin_amdgcn_wmma_f32_16x16x4_f32(false, A, false, B0, (short)0, acc0, false, false);
        acc1 = __builtin_amdgcn_wmma_f32_16x16x4_f32(false, A, false, B1, (short)0, acc1, false, false);
        acc2 = __builtin_amdgcn_wmma_f32_16x16x4_f32(false, A, false, B2, (short)0, acc2, false, false);
        acc3 = __builtin_amdgcn_wmma_f32_16x16x4_f32(false, A, false, B3, (short)0, acc3, false, false);
    }

    float* __restrict__ yb = y + ((size_t)b * 256 + m0) * NPIX;
    const int rh = hi * 8;
#pragma unroll
    for (int v = 0; v < 8; ++v) {
        const float bias = b_out[m0 + v + rh];
        float* dst = yb + (size_t)(v + rh) * NPIX + n0 + ln;
        dst[0]  = acc0[v] + bias;
        dst[16] = acc1[v] + bias;
        dst[32] = acc2[v] + bias;
        dst[48] = acc3[v] + bias;
    }
}

// ---------------------------------------------------------------------------
extern "C" void kernel_launch(void* const* d_in, const int* in_sizes, int n_in,
                              void* d_out, int out_size, void* d_ws, size_t ws_size,
                              hipStream_t stream) {
    (void)in_sizes; (void)n_in; (void)out_size; (void)ws_size;

    const float* x     = (const float*)d_in[0];   // (16,256,64,64)
    const float* w_qkv = (const float*)d_in[1];   // (384,256)
    const float* w_out = (const float*)d_in[2];   // (256,128)
    const float* b_out = (const float*)d_in[3];   // (256,)
    float*       y     = (float*)d_out;           // (16,256,64,64)

    // workspace carve-up (floats)
    float* qkv  = (float*)d_ws;                               // 16*384*4096
    float* rmax = qkv  + (size_t)NB * O3 * NPIX;              // 2048
    float* rsum = rmax + 2048;                                // 2048
    float* ctx  = rsum + 2048;                                // 16*4*32*32
    float* weff = ctx  + (size_t)NB * HEADS * 32 * 32;        // 16*256*128

    // 1) qkv GEMM: 16*24*64 waves / 8 waves per block
    qkv_gemm<<<3072, 256, 0, stream>>>(x, w_qkv, qkv);
    // 2) softmax stats: one block per k-row
    softmax_stats<<<2048, 256, 0, stream>>>(qkv, rmax, rsum);
    // 3) context: one wave per (b,h,dtile)
    context_wmma<<<128, 32, 0, stream>>>(qkv, rmax, rsum, ctx);
    // 4) fold w_out with ctx^T
    build_weff<<<2048, 256, 0, stream>>>(w_out, ctx, weff);
    // 5) fused output GEMM + bias: 16*16*64 waves / 8 per block
    out_gemm<<<2048, 256, 0, stream>>>(qkv, weff, b_out, y);
}